// SSIM_37340445672026
// MI455X (gfx1250) — compile-verified
//
#include <hip/hip_runtime.h>

typedef __attribute__((ext_vector_type(2))) float v2f;
typedef __attribute__((ext_vector_type(4))) float v4f;
typedef __attribute__((ext_vector_type(8))) float v8f;

#define H_IMG 512
#define W_IMG 512
#define PLANES 48           // 16 * 3
#define FW 28               // field buffer cols: 26 region + 2 K-pad
#define FH 32               // field buffer rows: 26 region + zero pad
#define HTW 32              // transposed-H cols (K dim, padded)
#define SSIM_C1 (0.01f * 0.01f)
#define SSIM_C2 (0.03f * 0.03f)

// 11-tap Gaussian (sigma=1.5), normalized; g[d]=0 outside [0,10].
__device__ __forceinline__ float gsel(int d) {
  float r = 0.0f;
  r = (d == 0)  ? 0.00102838f : r;
  r = (d == 1)  ? 0.00759876f : r;
  r = (d == 2)  ? 0.03600077f : r;
  r = (d == 3)  ? 0.10936070f : r;
  r = (d == 4)  ? 0.21300553f : r;
  r = (d == 5)  ? 0.26601172f : r;
  r = (d == 6)  ? 0.21300553f : r;
  r = (d == 7)  ? 0.10936070f : r;
  r = (d == 8)  ? 0.03600077f : r;
  r = (d == 9)  ? 0.00759876f : r;
  r = (d == 10) ? 0.00102838f : r;
  return r;
}

// Intra-wave LDS producer->consumer sync. All shared buffers below are
// per-wave private; CDNA5 LDS ops from one wave complete in order (DScnt),
// so we only need to stop the compiler from reordering across this point.
__device__ __forceinline__ void wave_lds_sync() {
  __builtin_amdgcn_fence(__ATOMIC_RELEASE, "wavefront");
  __builtin_amdgcn_wave_barrier();
  __builtin_amdgcn_fence(__ATOMIC_ACQUIRE, "wavefront");
}

__global__ __launch_bounds__(128)
void ssim_map_kernel(const float* __restrict__ img1,
                     const float* __restrict__ img2,
                     float* __restrict__ partial)
{
  __shared__ __align__(16) float sX [4][FH * FW];
  __shared__ __align__(16) float sY [4][FH * FW];
  __shared__ __align__(16) float sF [4][FH * FW];
  __shared__ __align__(16) float sHt[4][16 * HTW];
  __shared__ __align__(16) float sG [4][64];       // padded Gaussian lookup
  __shared__ float sRed[128];

  const int tid  = threadIdx.x;
  const int wave = tid >> 5;
  const int lane = tid & 31;
  const int hl   = lane >> 4;    // which 16-lane half
  const int l16  = lane & 15;

  // block -> plane + 32x32 tile; wave -> one 16x16 subtile
  const int bpp   = (W_IMG / 32) * (H_IMG / 32);   // 256 blocks/plane
  const int plane = blockIdx.x / bpp;
  const int bt    = blockIdx.x % bpp;
  const int row0  = (bt / (W_IMG / 32)) * 32 + ((wave >> 1) << 4);
  const int col0  = (bt % (W_IMG / 32)) * 32 + ((wave & 1) << 4);

  const float* base1 = img1 + (size_t)plane * (H_IMG * W_IMG);
  const float* base2 = img2 + (size_t)plane * (H_IMG * W_IMG);

  float* Xs  = sX[wave];
  float* Ys  = sY[wave];
  float* Fw  = sF[wave];
  float* Ht  = sHt[wave];
  float* Gw  = sG[wave];

  // Per-wave padded Gaussian table: Gw[16 + d] = g[d] (zeros elsewhere).
  for (int i = lane; i < 64; i += 32) Gw[i] = gsel(i - 16);

  // Stage 26x26 input regions (zero-padded to 32x28; SAME zero padding at borders)
  for (int idx = lane; idx < FH * FW; idx += 32) {
    const int rr = idx / FW, cc = idx % FW;
    const int gr = row0 - 5 + rr, gc = col0 - 5 + cc;
    float a = 0.0f, b = 0.0f;
    if (rr < 26 && cc < 26 && gr >= 0 && gr < H_IMG && gc >= 0 && gc < W_IMG) {
      const size_t off = (size_t)gr * W_IMG + gc;
      a = base1[off];
      b = base2[off];
    }
    Xs[idx] = a;
    Ys[idx] = b;
  }
  wave_lds_sync();

  // Banded Gaussian fragments from the LDS table.
  // f32 16x16x4 layout: lanes 0-15 hold K = 4*kk + {0,1}; lanes 16-31 K = 4*kk + {2,3}.
  // These fragments serve as B (horizontal pass) AND A (vertical pass).
  const float* gw = &Gw[16 + 2 * hl - l16];
  v2f band[7];
#pragma unroll
  for (int kk = 0; kk < 7; ++kk) {
    band[kk].x = gw[kk * 4];
    band[kk].y = gw[kk * 4 + 1];
  }

  v8f acc[5];
#pragma unroll
  for (int f = 0; f < 5; ++f) {
    const float* F = Xs;
    if (f == 1) F = Ys;
    if (f >= 2) {
      for (int idx = lane; idx < FH * FW; idx += 32) {
        const float a = Xs[idx], b = Ys[idx];
        Fw[idx] = (f == 2) ? a * a : (f == 3) ? b * b : a * b;
      }
      F = Fw;
      wave_lds_sync();
    }

    // ---- Horizontal pass: H(26x16) = F(26x28) x Band(28x16), two 16-row groups
#pragma unroll
    for (int grp = 0; grp < 2; ++grp) {
      v8f c = {0.f, 0.f, 0.f, 0.f, 0.f, 0.f, 0.f, 0.f};
#pragma unroll
      for (int kk = 0; kk < 7; ++kk) {
        const float* ap = &F[(grp * 16 + l16) * FW + kk * 4 + 2 * hl];
        v2f a; a.x = ap[0]; a.y = ap[1];
        c = __builtin_amdgcn_wmma_f32_16x16x4_f32(false, a, false, band[kk],
                                                  (short)0, c, false, false);
      }
      // D layout: vgpr r -> (M = r + 8*hl, N = l16); store transposed: Ht[N][M]
      float* hp = &Ht[l16 * HTW + grp * 16 + hl * 8];
      v4f lo = {c[0], c[1], c[2], c[3]};
      v4f hi = {c[4], c[5], c[6], c[7]};
      *(v4f*)(hp)     = lo;
      *(v4f*)(hp + 4) = hi;
    }
    wave_lds_sync();

    // ---- Vertical pass: V(16x16) = Band(16x28) x H(28x16); B read from Ht
    {
      v8f c = {0.f, 0.f, 0.f, 0.f, 0.f, 0.f, 0.f, 0.f};
#pragma unroll
      for (int kk = 0; kk < 7; ++kk) {
        const float* bp = &Ht[l16 * HTW + kk * 4 + 2 * hl];
        v2f b; b.x = bp[0]; b.y = bp[1];
        c = __builtin_amdgcn_wmma_f32_16x16x4_f32(false, band[kk], false, b,
                                                  (short)0, c, false, false);
      }
      acc[f] = c;
    }
    wave_lds_sync();   // Ht/Fw are rewritten next iteration
  }

  // Per-lane SSIM on 8 pixels (D layout: M = r + 8*hl, N = l16)
  float s = 0.0f;
#pragma unroll
  for (int r = 0; r < 8; ++r) {
    const float mu1 = acc[0][r], mu2 = acc[1][r];
    const float mu1sq = mu1 * mu1, mu2sq = mu2 * mu2, mu12 = mu1 * mu2;
    const float s1  = acc[2][r] - mu1sq;
    const float s2  = acc[3][r] - mu2sq;
    const float s12 = acc[4][r] - mu12;
    const float num = (2.0f * mu12 + SSIM_C1) * (2.0f * s12 + SSIM_C2);
    const float den = (mu1sq + mu2sq + SSIM_C1) * (s1 + s2 + SSIM_C2);
    s += num / den;
  }

  // Deterministic cross-wave block tree reduction (real barriers needed here)
  sRed[tid] = s;
  __syncthreads();
#pragma unroll
  for (int w = 64; w >= 1; w >>= 1) {
    if (tid < w) sRed[tid] += sRed[tid + w];
    __syncthreads();
  }
  if (tid == 0) partial[blockIdx.x] = sRed[0];
}

__global__ __launch_bounds__(256)
void ssim_reduce_kernel(const float* __restrict__ partial, int n,
                        float* __restrict__ out, float inv_count)
{
  __shared__ float red[256];
  const int tid = threadIdx.x;
  float s = 0.0f;
  for (int i = tid; i < n; i += 256) s += partial[i];   // fixed order: deterministic
  red[tid] = s;
  __syncthreads();
#pragma unroll
  for (int w = 128; w >= 1; w >>= 1) {
    if (tid < w) red[tid] += red[tid + w];
    __syncthreads();
  }
  if (tid == 0) out[0] = red[0] * inv_count;
}

extern "C" void kernel_launch(void* const* d_in, const int* in_sizes, int n_in,
                              void* d_out, int out_size, void* d_ws, size_t ws_size,
                              hipStream_t stream)
{
  (void)in_sizes; (void)n_in; (void)out_size; (void)ws_size;
  const float* img1 = (const float*)d_in[0];
  const float* img2 = (const float*)d_in[1];
  float* out     = (float*)d_out;
  float* partial = (float*)d_ws;

  const int nBlocks = PLANES * (W_IMG / 32) * (H_IMG / 32);   // 48 * 256 = 12288
  ssim_map_kernel<<<nBlocks, 128, 0, stream>>>(img1, img2, partial);

  const float inv_count = 1.0f / (float)((size_t)PLANES * H_IMG * W_IMG); // 1/12582912
  ssim_reduce_kernel<<<1, 256, 0, stream>>>(partial, nBlocks, out, inv_count);
}